// AttentionalProjection_26603027431840
// MI455X (gfx1250) — compile-verified
//
#include <hip/hip_runtime.h>
#include <cstdint>

// ---------------------------------------------------------------------------
// AttentionalProjection for MI455X (gfx1250, wave32).
//   S=1024, B=32, H=512, E=512, A=256
// Pipeline:
//   K1: eatt = emb @ W_enc^T + b_enc                (fp32 WMMA GEMM)
//   K2: qatt = outs[gather] @ W_dec^T + b_dec       (fp32 WMMA GEMM)
//   K3: scores -> masked softmax -> weights (d_out) (tanh/exp VALU kernel,
//       async-to-LDS double-buffered key streaming when available)
//   K4: ctx[:,b,:] = weights[:,:,b] @ emb[:,b,:]    (fp32 WMMA GEMM, per-b)
//   K5: out = outs@W_h2h^T + ctx@W_e2h^T + biases   (fp32 WMMA GEMM, fused K)
// ---------------------------------------------------------------------------

typedef __attribute__((ext_vector_type(2))) float v2f;
typedef __attribute__((ext_vector_type(8))) float v8f;
typedef __attribute__((ext_vector_type(4))) int   v4i;

constexpr int Sq = 1024;   // seq len
constexpr int Bc = 32;     // batch
constexpr int Hh = 512;    // hid dim
constexpr int Ee = 512;    // emb dim
constexpr int Aa = 256;    // att dim
constexpr int SB = Sq * Bc;

#define DEV_INLINE __device__ __forceinline__

#if defined(__has_builtin)
#if __has_builtin(__builtin_amdgcn_global_load_async_to_lds_b128) && \
    __has_builtin(__builtin_amdgcn_s_wait_asynccnt)
#define HAVE_ASYNC_LDS 1
#endif
#endif

#ifdef HAVE_ASYNC_LDS
// Builtin signature (from the toolchain's diagnostic):
//   param 0: v4i addrspace(1)* (global source)
//   param 1: v4i addrspace(3)* (LDS destination)
typedef __attribute__((address_space(1))) v4i as1_v4i;
typedef __attribute__((address_space(3))) v4i as3_v4i;
#endif

DEV_INLINE v8f wmma_f32(v2f a, v2f b, v8f c) {
  // V_WMMA_F32_16X16X4_F32 : D(16x16 f32) = A(16x4 f32) * B(4x16 f32) + C
  return __builtin_amdgcn_wmma_f32_16x16x4_f32(
      /*neg_a=*/false, a, /*neg_b=*/false, b,
      /*c_mod=*/(short)0, c, /*reuse_a=*/false, /*reuse_b=*/false);
}

DEV_INLINE float fast_tanhf(float x) {
  // tanh(x) = 1 - 2/(exp2(2x*log2(e)) + 1); saturates correctly at +-1.
  float e = __builtin_exp2f(x * 2.8853900817779268f);
  return 1.0f - 2.0f / (e + 1.0f);
}

DEV_INLINE float waveReduceSum(float v) {
#pragma unroll
  for (int off = 16; off > 0; off >>= 1) v += __shfl_xor(v, off, 32);
  return v;
}

DEV_INLINE float waveReduceMax(float v) {
#pragma unroll
  for (int off = 16; off > 0; off >>= 1) v = fmaxf(v, __shfl_xor(v, off, 32));
  return v;
}

// ---------------------------------------------------------------------------
// K1/K2: C[m,n] = sum_k Am[g(m),k] * W[n,k] + bias[n]
// Wave computes a 16(M) x 64(N) tile; 4 f32-WMMA accumulators reuse one A-frag.
// Grid: (M/16, N/256), block 128 (4 waves). GATHER maps row m -> m-B for m>=B.
// ---------------------------------------------------------------------------
template <bool GATHER>
__global__ void __launch_bounds__(128)
gemm_nt_bias(const float* __restrict__ Am, const float* __restrict__ W,
             const float* __restrict__ bias, float* __restrict__ C,
             int K, int N) {
  const int lane  = threadIdx.x & 31;
  const int wave  = threadIdx.x >> 5;
  const int m0    = blockIdx.x * 16;
  const int n0    = (blockIdx.y * 4 + wave) * 64;
  const int nlane = lane & 15;
  const int khalf = (lane >> 4) << 1;  // lanes 0-15 -> k+{0,1}, 16-31 -> k+{2,3}

  int mrow = m0 + nlane;
  if (GATHER) mrow = (mrow >= Bc) ? (mrow - Bc) : mrow;

  const float* arow = Am + (size_t)mrow * K;
  const float* wr   = W + (size_t)(n0 + nlane) * K;

  v8f acc[4] = {v8f{}, v8f{}, v8f{}, v8f{}};
  for (int k = 0; k < K; k += 4) {
    const int ko = k + khalf;
    v2f a = *(const v2f*)(arow + ko);
#pragma unroll
    for (int i = 0; i < 4; ++i) {
      v2f bf = *(const v2f*)(wr + (size_t)(16 * i) * K + ko);
      acc[i] = wmma_f32(a, bf, acc[i]);
    }
  }

  const int mbase = m0 + ((lane >> 4) << 3);  // C: vgpr j -> row j (+8 hi half)
#pragma unroll
  for (int i = 0; i < 4; ++i) {
    const int nn = n0 + 16 * i + nlane;
    const float bb = bias[nn];
#pragma unroll
    for (int j = 0; j < 8; ++j)
      C[(size_t)(mbase + j) * N + nn] = acc[i][j] + bb;
  }
}

// ---------------------------------------------------------------------------
// K3: per-(q,b) workgroup (256 threads = 8 waves).
// score[k] = sum_a v[a]*tanh(qatt[q,b,a]+eatt[k,b,a]) for k < L=max(1,q-1),
// then masked softmax over k, weights written to wts[q,k,b] (zeros for k>=L).
// Each wave owns one key per iteration; each lane owns 8 of the 256 A-dims.
// When the toolchain exposes GLOBAL_LOAD_ASYNC_TO_LDS, each wave
// double-buffers its next key row (1 KB) into LDS while computing the
// current one, synchronizing with s_wait_asynccnt (ASYNCcnt).
// ---------------------------------------------------------------------------
__global__ void __launch_bounds__(256)
scores_softmax_kernel(const float* __restrict__ qatt,
                      const float* __restrict__ eatt,
                      const float* __restrict__ vvec,
                      float* __restrict__ wts) {
  const int q = blockIdx.x, b = blockIdx.y;
  const int tid = threadIdx.x, lane = tid & 31, wave = tid >> 5;
  int L = q - 1;
  if (L < 1) L = 1;

  __shared__ float sc[Sq];
  __shared__ float red[8];

  const float4* qp = (const float4*)(qatt + ((size_t)q * Bc + b) * Aa + lane * 8);
  const float4  qa0 = qp[0], qa1 = qp[1];
  const float4* vp = (const float4*)(vvec + lane * 8);
  const float4  v0 = vp[0], v1 = vp[1];

#ifdef HAVE_ASYNC_LDS
  // [wave][buf][a] : per-wave private double buffer, 16 KB total.
  __shared__ __align__(16) float ebuf[8][2][Aa];
  {
    auto issue_copy = [&](int k, int bufIdx) {
      // Each lane copies its own 32 B (a-range lane*8..lane*8+7) as 2 x b128.
      const float* er = eatt + ((size_t)k * Bc + b) * Aa + lane * 8;
      float* dp = &ebuf[wave][bufIdx][lane * 8];
      __builtin_amdgcn_global_load_async_to_lds_b128(
          (as1_v4i*)er, (as3_v4i*)dp, 0, 0);
      __builtin_amdgcn_global_load_async_to_lds_b128(
          (as1_v4i*)(er + 4), (as3_v4i*)(dp + 4), 0, 0);
    };
    int bufIdx = 0;
    int k = wave;
    if (k < L) issue_copy(k, 0);
    for (; k < L; k += 8) {
      const int knext = k + 8;
      if (knext < L) {
        issue_copy(knext, bufIdx ^ 1);                // prefetch next key row
        __builtin_amdgcn_s_wait_asynccnt(2);          // current row resident
      } else {
        __builtin_amdgcn_s_wait_asynccnt(0);
      }
      const float4* ep = (const float4*)&ebuf[wave][bufIdx][lane * 8];
      const float4 e0 = ep[0], e1 = ep[1];
      float acc = v0.x * fast_tanhf(qa0.x + e0.x) +
                  v0.y * fast_tanhf(qa0.y + e0.y) +
                  v0.z * fast_tanhf(qa0.z + e0.z) +
                  v0.w * fast_tanhf(qa0.w + e0.w) +
                  v1.x * fast_tanhf(qa1.x + e1.x) +
                  v1.y * fast_tanhf(qa1.y + e1.y) +
                  v1.z * fast_tanhf(qa1.z + e1.z) +
                  v1.w * fast_tanhf(qa1.w + e1.w);
      acc = waveReduceSum(acc);
      if (lane == 0) sc[k] = acc;
      bufIdx ^= 1;
    }
  }
#else
  for (int k = wave; k < Sq; k += 8) {
    if (k < L) {
      const float* er = eatt + ((size_t)k * Bc + b) * Aa + lane * 8;
      if (k + 8 < L)  // stream the next key row this wave will touch
        __builtin_prefetch(er + (size_t)8 * Bc * Aa, 0, 3);
      const float4 e0 = ((const float4*)er)[0];
      const float4 e1 = ((const float4*)er)[1];
      float acc = v0.x * fast_tanhf(qa0.x + e0.x) +
                  v0.y * fast_tanhf(qa0.y + e0.y) +
                  v0.z * fast_tanhf(qa0.z + e0.z) +
                  v0.w * fast_tanhf(qa0.w + e0.w) +
                  v1.x * fast_tanhf(qa1.x + e1.x) +
                  v1.y * fast_tanhf(qa1.y + e1.y) +
                  v1.z * fast_tanhf(qa1.z + e1.z) +
                  v1.w * fast_tanhf(qa1.w + e1.w);
      acc = waveReduceSum(acc);
      if (lane == 0) sc[k] = acc;
    }
  }
#endif
  __syncthreads();

  // block max over valid keys
  float mx = -3.4e38f;
  for (int k = tid; k < L; k += 256) mx = fmaxf(mx, sc[k]);
  mx = waveReduceMax(mx);
  if (lane == 0) red[wave] = mx;
  __syncthreads();
  float m2 = red[0];
#pragma unroll
  for (int i = 1; i < 8; ++i) m2 = fmaxf(m2, red[i]);
  __syncthreads();  // red reused below

  // exp + block sum (each thread re-reads only entries it wrote)
  float sum = 0.f;
  for (int k = tid; k < L; k += 256) {
    float e = __builtin_exp2f((sc[k] - m2) * 1.4426950408889634f);
    sum += e;
    sc[k] = e;
  }
  sum = waveReduceSum(sum);
  if (lane == 0) red[wave] = sum;
  __syncthreads();
  float tot = red[0];
#pragma unroll
  for (int i = 1; i < 8; ++i) tot += red[i];
  const float inv = 1.0f / tot;

  const size_t base = (size_t)q * Sq * Bc + b;
  for (int k = tid; k < Sq; k += 256) {
    const float w = (k < L) ? sc[k] * inv : 0.f;
    wts[base + (size_t)k * Bc] = w;
  }
}

// ---------------------------------------------------------------------------
// K4: ctx[q,b,e] = sum_k wts[q,k,b] * emb[k,b,e]   (per-b GEMM, fp32 WMMA)
// Grid: (Sq/16, Ee/256, Bc), block 128. K clipped to the live (masked) range.
// ---------------------------------------------------------------------------
__global__ void __launch_bounds__(128)
context_gemm_kernel(const float* __restrict__ wts, const float* __restrict__ emb,
                    float* __restrict__ ctx) {
  const int lane  = threadIdx.x & 31;
  const int wave  = threadIdx.x >> 5;
  const int q0    = blockIdx.x * 16;
  const int e0    = (blockIdx.y * 4 + wave) * 64;
  const int b     = blockIdx.z;
  const int elane = lane & 15;
  const int khalf = (lane >> 4) << 1;
  const int qrow  = q0 + elane;

  int kEnd = q0 + 14;                 // max L over this q-tile
  if (kEnd < 1) kEnd = 1;
  kEnd = (kEnd + 3) & ~3;
  if (kEnd > Sq) kEnd = Sq;

  const size_t wbase = (size_t)qrow * Sq * Bc + b;
  v8f acc[4] = {v8f{}, v8f{}, v8f{}, v8f{}};

  for (int k = 0; k < kEnd; k += 4) {
    const int ko = k + khalf;
    v2f a;
    a[0] = wts[wbase + (size_t)ko * Bc];
    a[1] = wts[wbase + (size_t)(ko + 1) * Bc];
    const float* eb0 = emb + ((size_t)ko * Bc + b) * Ee + e0 + elane;
    const float* eb1 = eb0 + (size_t)Bc * Ee;  // k+1 row
#pragma unroll
    for (int i = 0; i < 4; ++i) {
      v2f bf;
      bf[0] = eb0[16 * i];
      bf[1] = eb1[16 * i];
      acc[i] = wmma_f32(a, bf, acc[i]);
    }
  }

  const int mb = q0 + ((lane >> 4) << 3);
#pragma unroll
  for (int i = 0; i < 4; ++i) {
    const int e = e0 + 16 * i + elane;
#pragma unroll
    for (int j = 0; j < 8; ++j)
      ctx[((size_t)(mb + j) * Bc + b) * Ee + e] = acc[i][j];
  }
}

// ---------------------------------------------------------------------------
// K5: out[m,n] = sum_k outs[m,k]W_h2h[n,k] + sum_k ctx[m,k]W_e2h[n,k] + biases
// Grid: (SB/16, Hh/256), block 128.
// ---------------------------------------------------------------------------
__global__ void __launch_bounds__(128)
fuse_out_kernel(const float* __restrict__ X1, const float* __restrict__ W1,
                const float* __restrict__ X2, const float* __restrict__ W2,
                const float* __restrict__ b1, const float* __restrict__ b2,
                float* __restrict__ Out) {
  const int lane  = threadIdx.x & 31;
  const int wave  = threadIdx.x >> 5;
  const int m0    = blockIdx.x * 16;
  const int n0    = (blockIdx.y * 4 + wave) * 64;
  const int nlane = lane & 15;
  const int khalf = (lane >> 4) << 1;
  const int mrow  = m0 + nlane;

  v8f acc[4] = {v8f{}, v8f{}, v8f{}, v8f{}};

  {
    const float* arow = X1 + (size_t)mrow * Hh;
    const float* wr   = W1 + (size_t)(n0 + nlane) * Hh;
    for (int k = 0; k < Hh; k += 4) {
      const int ko = k + khalf;
      v2f a = *(const v2f*)(arow + ko);
#pragma unroll
      for (int i = 0; i < 4; ++i) {
        v2f bf = *(const v2f*)(wr + (size_t)(16 * i) * Hh + ko);
        acc[i] = wmma_f32(a, bf, acc[i]);
      }
    }
  }
  {
    const float* arow = X2 + (size_t)mrow * Ee;
    const float* wr   = W2 + (size_t)(n0 + nlane) * Ee;
    for (int k = 0; k < Ee; k += 4) {
      const int ko = k + khalf;
      v2f a = *(const v2f*)(arow + ko);
#pragma unroll
      for (int i = 0; i < 4; ++i) {
        v2f bf = *(const v2f*)(wr + (size_t)(16 * i) * Ee + ko);
        acc[i] = wmma_f32(a, bf, acc[i]);
      }
    }
  }

  const int mbase = m0 + ((lane >> 4) << 3);
#pragma unroll
  for (int i = 0; i < 4; ++i) {
    const int nn = n0 + 16 * i + nlane;
    const float bb = b1[nn] + b2[nn];
#pragma unroll
    for (int j = 0; j < 8; ++j)
      Out[(size_t)(mbase + j) * Hh + nn] = acc[i][j] + bb;
  }
}

// ---------------------------------------------------------------------------
extern "C" void kernel_launch(void* const* d_in, const int* in_sizes, int n_in,
                              void* d_out, int out_size, void* d_ws, size_t ws_size,
                              hipStream_t stream) {
  const float* outs  = (const float*)d_in[0];
  const float* emb   = (const float*)d_in[1];
  const float* W_enc = (const float*)d_in[2];
  const float* b_enc = (const float*)d_in[3];
  const float* W_dec = (const float*)d_in[4];
  const float* b_dec = (const float*)d_in[5];
  const float* vvec  = (const float*)d_in[6];
  const float* W_h2h = (const float*)d_in[7];
  const float* b_h2h = (const float*)d_in[8];
  const float* W_e2h = (const float*)d_in[9];
  const float* b_e2h = (const float*)d_in[10];

  float* out = (float*)d_out;                       // (S,B,H)
  float* wts = out + (size_t)SB * Hh;               // (S,S,B)

  float* eatt = (float*)d_ws;                       // (S,B,A)  32 MB
  float* qatt = eatt + (size_t)SB * Aa;             // (S,B,A)  32 MB
  float* ctx  = qatt + (size_t)SB * Aa;             // (S,B,E)  67 MB

  // K1: emb_att = emb @ W_enc^T + b_enc
  gemm_nt_bias<false><<<dim3(SB / 16, Aa / 256), 128, 0, stream>>>(
      emb, W_enc, b_enc, eatt, Ee, Aa);
  // K2: q_att = outs[max(0,s-1)] @ W_dec^T + b_dec  (row gather: m>=B -> m-B)
  gemm_nt_bias<true><<<dim3(SB / 16, Aa / 256), 128, 0, stream>>>(
      outs, W_dec, b_dec, qatt, Hh, Aa);
  // K3: masked additive-attention scores + softmax -> weights (into d_out)
  scores_softmax_kernel<<<dim3(Sq, Bc), 256, 0, stream>>>(qatt, eatt, vvec, wts);
  // K4: context = weights @ emb (per batch)
  context_gemm_kernel<<<dim3(Sq / 16, Ee / 256, Bc), 128, 0, stream>>>(
      wts, emb, ctx);
  // K5: output = outs@W_h2h^T + context@W_e2h^T + biases
  fuse_out_kernel<<<dim3(SB / 16, Hh / 256), 128, 0, stream>>>(
      outs, W_h2h, ctx, W_e2h, b_h2h, b_e2h, out);
}